// CorrLossChamfer_63771674411371
// MI455X (gfx1250) — compile-verified
//
#include <hip/hip_runtime.h>
#include <cstdint>
#include <cstddef>

typedef __attribute__((ext_vector_type(2))) float v2f;
typedef __attribute__((ext_vector_type(8))) float v8f;

#define BATCH   16
#define M_PTS   2048
#define NVERTS  8000
#define TOTAL_J 6000           // 2000 head + 2000 belly + 1000 neck + 1000 back
#define TILES_PER_B 376        // 125 + 125 + 63 + 63
#define TOT_TILES (BATCH * TILES_PER_B)   // 6016

// ---------------------------------------------------------------------------
// Stage 1: gather + quaternion rotate + weak-perspective project.
// Writes vert2d to d_out[1 + 2*(b*6000 + j) + c]  (reference output layout).
// ---------------------------------------------------------------------------
__global__ void proj_kernel(const float* __restrict__ verts,
                            const float* __restrict__ cams,
                            const int* __restrict__ hidx,
                            const int* __restrict__ bidx,
                            const int* __restrict__ nidx,
                            const int* __restrict__ kidx,
                            float* __restrict__ out)
{
    int gid = blockIdx.x * blockDim.x + threadIdx.x;
    if (gid >= BATCH * TOTAL_J) return;
    int b = gid / TOTAL_J;
    int j = gid - b * TOTAL_J;

    int vidx;
    if (j < 2000)      vidx = hidx[j];
    else if (j < 4000) vidx = bidx[j - 2000];
    else if (j < 5000) vidx = nidx[j - 4000];
    else               vidx = kidx[j - 5000];

    const float* c = cams + b * 7;
    float s = c[0], tx = c[1], ty = c[2];
    float qw = c[3], qx = c[4], qy = c[5], qz = c[6];
    float inv = 1.0f / (sqrtf(qw*qw + qx*qx + qy*qy + qz*qz) + 1e-8f);
    qw *= inv; qx *= inv; qy *= inv; qz *= inv;

    const float* v = verts + ((size_t)b * NVERTS + (size_t)vidx) * 3;
    float X = v[0], Y = v[1], Z = v[2];

    float r00 = 1.0f - 2.0f*(qy*qy + qz*qz);
    float r01 = 2.0f*(qx*qy - qw*qz);
    float r02 = 2.0f*(qx*qz + qw*qy);
    float r10 = 2.0f*(qx*qy + qw*qz);
    float r11 = 1.0f - 2.0f*(qx*qx + qz*qz);
    float r12 = 2.0f*(qy*qz - qw*qx);

    float px = s * (r00*X + r01*Y + r02*Z) + tx;
    float py = s * (r10*X + r11*Y + r12*Z) + ty;

    out[1 + 2*gid]     = px;
    out[1 + 2*gid + 1] = py;
}

// ---------------------------------------------------------------------------
// Stage 2: chamfer via V_WMMA_F32_16X16X4_F32.
// A row  = [x, y, 1, x^2+y^2]          (16 projected vertices)
// B col  = [-2px, -2py, px^2+py^2, 1]  (16 target points)
// A . B  = full squared distance. One wave = one 16-row tile, 128 K-chunks,
// running elementwise min on the 8 accumulator VGPRs, then shfl-xor min tree.
// ---------------------------------------------------------------------------
__global__ void chamfer_kernel(const float* __restrict__ v2d,   // d_out + 1
                               const float* __restrict__ hp,
                               const float* __restrict__ bp,
                               const float* __restrict__ np,
                               const float* __restrict__ kp,
                               float* __restrict__ partial)
{
    int wave = blockIdx.x * (blockDim.x >> 5) + (threadIdx.x >> 5);
    if (wave >= TOT_TILES) return;                 // wave-uniform guard

    int b = wave / TILES_PER_B;
    int t = wave - b * TILES_PER_B;

    int tloc, colOff, Npart;
    const float* pts;
    float w;
    if (t < 125)      { tloc = t;       colOff = 0;    Npart = 2000; pts = hp; w = 0.45f; }
    else if (t < 250) { tloc = t - 125; colOff = 2000; Npart = 2000; pts = bp; w = 0.45f; }
    else if (t < 313) { tloc = t - 250; colOff = 4000; Npart = 1000; pts = np; w = 0.05f; }
    else              { tloc = t - 313; colOff = 5000; Npart = 1000; pts = kp; w = 0.05f; }

    int rowBase = tloc * 16;
    int lane = threadIdx.x & 31;
    int r16  = lane & 15;
    bool hi  = lane >= 16;

    // Build loop-invariant A fragment (16x4 f32: V0=K0/K2, V1=K1/K3 per lane half).
    // d_out+1 is only 4B-aligned, so these two loads stay scalar b32.
    int row = rowBase + r16;
    float x = 0.0f, y = 0.0f;
    if (row < Npart) {
        const float* p = v2d + ((size_t)b * TOTAL_J + colOff + row) * 2;
        x = p[0];
        y = p[1];
    }
    v2f A;
    A.x = hi ? 1.0f        : x;   // K2 : K0
    A.y = hi ? (x*x + y*y) : y;   // K3 : K1

    v8f rmin;
    #pragma unroll
    for (int r = 0; r < 8; ++r) rmin[r] = 3.0e38f;

    // Target points: 8B-aligned pairs -> vectorized global_load_b64 per chunk.
    const v2f* pbase = (const v2f*)(pts + (size_t)b * M_PTS * 2);
    v8f czero = {};

    #pragma unroll 4
    for (int c = 0; c < M_PTS / 16; ++c) {
        v2f pq = pbase[c * 16 + r16];     // {px, py} — one b64 load
        float px = pq.x, py = pq.y;
        v2f Bf;
        Bf.x = hi ? (px*px + py*py) : (-2.0f * px);  // K2 : K0
        Bf.y = hi ? 1.0f            : (-2.0f * py);  // K3 : K1

        v8f d = __builtin_amdgcn_wmma_f32_16x16x4_f32(
            /*neg_a=*/false, A, /*neg_b=*/false, Bf,
            /*c_mod=*/(short)0, czero, /*reuse_a=*/false, /*reuse_b=*/false);

        #pragma unroll
        for (int r = 0; r < 8; ++r) rmin[r] = fminf(rmin[r], d[r]);
    }

    // Min across each 16-lane half (xor offsets < 16 stay within the half),
    // then lanes 0 / 16 hold the minima for rows r and r+8.
    float sum = 0.0f;
    #pragma unroll
    for (int r = 0; r < 8; ++r) {
        float v = rmin[r];
        v = fminf(v, __shfl_xor(v, 8, 32));
        v = fminf(v, __shfl_xor(v, 4, 32));
        v = fminf(v, __shfl_xor(v, 2, 32));
        v = fminf(v, __shfl_xor(v, 1, 32));
        int rg = rowBase + r + (hi ? 8 : 0);
        if ((lane == 0 || lane == 16) && rg < Npart) sum += v;
    }
    sum += __shfl_xor(sum, 16, 32);
    if (lane == 0)
        partial[wave] = sum * (w / (float)(BATCH * TOTAL_J));
}

// ---------------------------------------------------------------------------
// Stage 3: deterministic fixed-tree reduction of per-tile partials -> loss.
// ---------------------------------------------------------------------------
__global__ void reduce_kernel(const float* __restrict__ partial, int n,
                              float* __restrict__ out)
{
    __shared__ float sh[256];
    float s = 0.0f;
    for (int i = threadIdx.x; i < n; i += 256) s += partial[i];
    sh[threadIdx.x] = s;
    __syncthreads();
    for (int off = 128; off > 0; off >>= 1) {
        if ((int)threadIdx.x < off) sh[threadIdx.x] += sh[threadIdx.x + off];
        __syncthreads();
    }
    if (threadIdx.x == 0) out[0] = sh[0];
}

// ---------------------------------------------------------------------------
extern "C" void kernel_launch(void* const* d_in, const int* in_sizes, int n_in,
                              void* d_out, int out_size, void* d_ws, size_t ws_size,
                              hipStream_t stream)
{
    const float* hp    = (const float*)d_in[0];
    const float* bp    = (const float*)d_in[1];
    const float* np_   = (const float*)d_in[2];
    const float* kp    = (const float*)d_in[3];
    const float* verts = (const float*)d_in[4];
    const float* cams  = (const float*)d_in[5];
    const int*   hidx  = (const int*)d_in[6];
    const int*   bidx  = (const int*)d_in[7];
    const int*   nidx  = (const int*)d_in[8];
    const int*   kidx  = (const int*)d_in[9];

    float* out     = (float*)d_out;
    float* partial = (float*)d_ws;          // 6016 floats (~24 KB)

    // Stage 1: projection (96000 points)
    proj_kernel<<<(BATCH * TOTAL_J + 255) / 256, 256, 0, stream>>>(
        verts, cams, hidx, bidx, nidx, kidx, out);

    // Stage 2: WMMA chamfer, 6016 tiles, 8 waves per 256-thread block
    chamfer_kernel<<<TOT_TILES / 8, 256, 0, stream>>>(
        out + 1, hp, bp, np_, kp, partial);

    // Stage 3: deterministic loss reduction
    reduce_kernel<<<1, 256, 0, stream>>>(partial, TOT_TILES, out);
}